// MultiheadCosAttention_32246614459227
// MI455X (gfx1250) — compile-verified
//
#include <hip/hip_runtime.h>

#define TDIM 2048
#define SDIM 2048
#define BDIM 4
#define EDIM 1024
#define NCH  8
#define SC   (SDIM / NCH)     /* 256 */
#define EPSV 1e-8f

/* workspace layout, in floats */
#define OFF_CQ    0
#define OFF_SQ    (OFF_CQ + TDIM)              /* 2048  */
#define OFF_DINV  (OFF_SQ + TDIM)              /* 4096  */
#define OFF_R     (OFF_DINV + TDIM)            /* 6144  : 16 x EDIM (rows 8..15 zero) */
#define OFF_RPART (OFF_R + 16 * EDIM)          /* 22528 : NCH x 8 x EDIM */
#define OFF_P     (OFF_RPART + NCH * 8 * EDIM) /* 88064 : EDIM x 16 */

typedef __attribute__((ext_vector_type(2))) float v2f;
typedef __attribute__((ext_vector_type(8))) float v8f;

/* ---- K1: trig tables (aq == ak since T == S == MAX_L) + zero R ---- */
__global__ void k_init(float* __restrict__ ws) {
    int i = blockIdx.x * blockDim.x + threadIdx.x;
    if (i < TDIM) {
        float a = 1.5707963267948966f * (float)(i + 1) * (1.0f / 2048.0f);
        ws[OFF_CQ + i] = cosf(a);
        ws[OFF_SQ + i] = sinf(a);
    }
    if (i < 16 * EDIM) ws[OFF_R + i] = 0.0f;
}

/* ---- K2: partial reductions r_c/r_s over s-chunks of `value` (S,B,E) ---- */
__global__ __launch_bounds__(128) void k_rpart(const float* __restrict__ value,
                                               float* __restrict__ ws) {
    __shared__ float sc[SC], ss[SC];
    const int tid = threadIdx.x;
    const int b = blockIdx.y, ch = blockIdx.z;
    const int e = blockIdx.x * 128 + tid;
    for (int i = tid; i < SC; i += 128) {
        sc[i] = ws[OFF_CQ + ch * SC + i];
        ss[i] = ws[OFF_SQ + ch * SC + i];
    }
    __syncthreads();
    float ac = 0.0f, as = 0.0f;
    const float* vptr = value + (size_t)(ch * SC) * (BDIM * EDIM) + (size_t)b * EDIM + e;
    for (int s = 0; s < SC; ++s) {
        float vv = vptr[(size_t)s * (BDIM * EDIM)];
        ac = fmaf(sc[s], vv, ac);
        as = fmaf(ss[s], vv, as);
    }
    float* rp = ws + OFF_RPART + ((size_t)ch * 8 + 2 * b) * EDIM + e;
    rp[0]    = ac;
    rp[EDIM] = as;
}

/* ---- K2b: sum partials into R rows 0..7 ---- */
__global__ void k_rsum(float* __restrict__ ws) {
    int idx = blockIdx.x * 256 + threadIdx.x;   /* 0 .. 8*EDIM-1 */
    if (idx >= 8 * EDIM) return;
    float s = 0.0f;
    for (int ch = 0; ch < NCH; ++ch) s += ws[OFF_RPART + ch * 8 * EDIM + idx];
    ws[OFF_R + idx] = s;
}

/* ---- K3: row L1 norms D[t], write normalized w, store 1/max(D,eps) ---- */
__global__ __launch_bounds__(256) void k_w(float* __restrict__ ws, float* __restrict__ wout) {
    __shared__ float s_cq[SDIM];
    __shared__ float s_sq[SDIM];
    __shared__ float red[256];
    __shared__ float inv_s;
    const int tid = threadIdx.x;
    const int t = blockIdx.x;
    for (int i = tid; i < SDIM; i += 256) {
        s_cq[i] = ws[OFF_CQ + i];
        s_sq[i] = ws[OFF_SQ + i];
    }
    __syncthreads();
    const float ct = s_cq[t], st = s_sq[t];
    float p = 0.0f;
    for (int s = tid; s < SDIM; s += 256)
        p += fabsf(fmaf(ct, s_cq[s], st * s_sq[s]));
    red[tid] = p;
    __syncthreads();
    for (int off = 128; off > 0; off >>= 1) {
        if (tid < off) red[tid] += red[tid + off];
        __syncthreads();
    }
    if (tid == 0) {
        float inv = 1.0f / fmaxf(red[0], EPSV);
        inv_s = inv;
        ws[OFF_DINV + t] = inv;
    }
    __syncthreads();
    const float inv = inv_s;
    float* wrow = wout + (size_t)t * SDIM;
    for (int s = tid; s < SDIM; s += 256)
        wrow[s] = fmaf(ct, s_cq[s], st * s_sq[s]) * inv;
}

/* ---- K4: P[e,j] = sum_k v_w[e,k] * R[j,k]  via V_WMMA_F32_16X16X4_F32 ----
 * One wave per 16-row M tile of v_w.  A: 16x4 fp32 (v_w rows), B: 4x16 fp32
 * (R rows as columns, rows 8..15 zero).  Lane L holds M/N = L&15 with
 * K-offset 2*(L>>4); VGPR index walks K.  256 accumulating WMMAs (K=1024). */
__global__ __launch_bounds__(32) void k_gemm(const float* __restrict__ vw,
                                             const float* __restrict__ R,
                                             float* __restrict__ P) {
    const int lane = threadIdx.x;
    const int e0   = blockIdx.x * 16;
    const int m    = lane & 15;
    const int koff = (lane >> 4) * 2;
    const float* arow = vw + (size_t)(e0 + m) * EDIM + koff;
    const float* brow = R  + (size_t)m * EDIM + koff;
    v8f acc = {};
#pragma unroll 4
    for (int k = 0; k < EDIM; k += 4) {
        v2f a = *(const v2f*)(arow + k);
        v2f b = *(const v2f*)(brow + k);
        acc = __builtin_amdgcn_wmma_f32_16x16x4_f32(
            /*neg_a=*/false, a, /*neg_b=*/false, b,
            /*c_mod=*/(short)0, acc, /*reuse_a=*/false, /*reuse_b=*/false);
    }
    /* D layout: VGPR r, lanes 0-15 -> M=r, lanes 16-31 -> M=r+8, N = lane&15 */
    const int mb = (lane >> 4) * 8;
#pragma unroll
    for (int r = 0; r < 8; ++r)
        P[(size_t)(e0 + mb + r) * 16 + m] = acc[r];
}

/* ---- K5: out[t,b,e] = inv[t]*(cq[t]*P[e,2b] + sq[t]*P[e,2b+1]) + v_b[e] ---- */
__global__ __launch_bounds__(256) void k_out(const float* __restrict__ ws,
                                             const float* __restrict__ vb,
                                             float* __restrict__ out) {
    const int tid = threadIdx.x;
    const int t = blockIdx.x / BDIM;
    const int b = blockIdx.x % BDIM;
    const float inv = ws[OFF_DINV + t];
    const float ct = ws[OFF_CQ + t] * inv;
    const float st = ws[OFF_SQ + t] * inv;
    const float* P = ws + OFF_P;
    float* o = out + ((size_t)t * BDIM + b) * EDIM;
    for (int e = tid; e < EDIM; e += 256) {
        float pc = P[(size_t)e * 16 + 2 * b];
        float ps = P[(size_t)e * 16 + 2 * b + 1];
        o[e] = fmaf(ct, pc, fmaf(st, ps, vb[e]));
    }
}

extern "C" void kernel_launch(void* const* d_in, const int* in_sizes, int n_in,
                              void* d_out, int out_size, void* d_ws, size_t ws_size,
                              hipStream_t stream) {
    (void)in_sizes; (void)n_in; (void)out_size; (void)ws_size;
    /* inputs: 0=query (unused), 1=key (unused), 2=value, 3=v_w, 4=v_b */
    const float* value = (const float*)d_in[2];
    const float* vw    = (const float*)d_in[3];
    const float* vb    = (const float*)d_in[4];
    float* out  = (float*)d_out;                         /* (T,B,E) */
    float* wout = out + (size_t)TDIM * BDIM * EDIM;      /* (1,T,S) */
    float* ws   = (float*)d_ws;

    k_init <<<dim3((16 * EDIM + 255) / 256), 256, 0, stream>>>(ws);
    k_rpart<<<dim3(EDIM / 128, BDIM, NCH),   128, 0, stream>>>(value, ws);
    k_rsum <<<dim3(8 * EDIM / 256),          256, 0, stream>>>(ws);
    k_w    <<<dim3(TDIM),                    256, 0, stream>>>(ws, wout);
    k_gemm <<<dim3(EDIM / 16),                32, 0, stream>>>(vw, ws + OFF_R, ws + OFF_P);
    k_out  <<<dim3(TDIM * BDIM),             256, 0, stream>>>(ws, vb, out);
}